// RNNGNNLayer_22909355557044
// MI455X (gfx1250) — compile-verified
//
#include <hip/hip_runtime.h>
#include <stdint.h>

// ---------------- tunables ----------------
#define WAVES        8                       // waves per block (wave32)
#define TBLOCK       (WAVES * 32)            // 256 threads
#define PIPE         8                       // ring depth (power of 2)
#define EPT          16                      // f32 outputs per thread (4x float4)
#define WAVE_FLOATS  (32 * EPT)              // 512 floats per wave strip
#define BLOCK_FLOATS (WAVES * WAVE_FLOATS)   // 4096 floats per block
#define BUF_BYTES    (WAVE_FLOATS * 4)       // 2048 B per LDS buffer

// Async DMA global -> LDS, 16B per lane. Per CDNA5 ISA (GV mode):
//   16 B at global (vaddr + offset) are copied to LDS (vdst + offset)
// IOFFSET is applied to BOTH addresses, so one base pair serves all q-chunks.
__device__ __forceinline__ void async_b128(uint32_t lds_addr, const float* g, int imm) {
    asm volatile("global_load_async_to_lds_b128 %0, %1, off offset:%2"
                 :: "v"(lds_addr), "v"(g), "n"(imm)
                 : "memory");
}

template <int N>
__device__ __forceinline__ void wait_async() {
    asm volatile("s_wait_asynccnt %0" :: "n"(N) : "memory");
}

__device__ __forceinline__ void wait_ds0() {
    asm volatile("s_wait_dscnt 0" ::: "memory");
}

__global__ void __launch_bounds__(TBLOCK, 1)
ewma_fold_kernel(const float* __restrict__ adj,
                 const float* __restrict__ lam_p,
                 float* __restrict__ out,
                 int T, int NN)
{
    extern __shared__ float smem[];   // WAVES * PIPE * 512 floats = 128 KB

    const int wave = threadIdx.x >> 5;
    const int lane = threadIdx.x & 31;

    float lam = lam_p[0];
    lam = fminf(fmaxf(lam, 0.0f), 1.0f);     // jnp.clip(lam, 0, 1)
    const float om = 1.0f - lam;

    // This wave's 512-float strip of the output / of each slice.
    const size_t segBase = (size_t)blockIdx.x * BLOCK_FLOATS
                         + (size_t)wave * WAVE_FLOATS;
    const float* gbase = adj + segBase + (size_t)(lane * 4);

    // Raw 32-bit LDS address (low 32 bits of generic pointer == LDS offset).
    const uint32_t ldsLane = (uint32_t)(uintptr_t)smem
                           + (uint32_t)(wave * PIPE * BUF_BYTES)
                           + (uint32_t)(lane * 16);

    // Issue one slice (4 x b128 per lane = 2 KB per wave) into ring buffer `buf`.
    auto issue = [&](int t, int buf) {
        const float*   g = gbase + (size_t)t * (size_t)NN;
        const uint32_t l = ldsLane + (uint32_t)(buf * BUF_BYTES);
        async_b128(l, g, 0 * 512);
        async_b128(l, g, 1 * 512);
        async_b128(l, g, 2 * 512);
        async_b128(l, g, 3 * 512);
    };

    float4 acc[4];

    // Consume slice from ring buffer `buf`: acc = (1-lam)*acc + lam*x
    auto consume = [&](int buf, bool first) {
        const int base = (wave * PIPE + buf) * WAVE_FLOATS + lane * 4;
#pragma unroll
        for (int q = 0; q < 4; ++q) {
            const float4 x = *reinterpret_cast<const float4*>(&smem[base + q * 128]);
            if (first) {
                acc[q] = x;                           // acc = adj[0]
            } else {
                acc[q].x = fmaf(om, acc[q].x, lam * x.x);
                acc[q].y = fmaf(om, acc[q].y, lam * x.y);
                acc[q].z = fmaf(om, acc[q].z, lam * x.z);
                acc[q].w = fmaf(om, acc[q].w, lam * x.w);
            }
        }
    };

    // ---- prologue: fill the ring (slices 0..PIPE-1) ----
#pragma unroll
    for (int p = 0; p < PIPE; ++p) issue(p, p);

    // ---- t = 0 (initializes acc) ----
    // Outstanding = PIPE slices * 4 ops = 32; oldest slice done when cnt <= 28.
    wait_async<(PIPE - 1) * 4>();
    consume(0, true);
    wait_ds0();                 // LDS reads landed before we overwrite the buffer
    issue(PIPE, 0);

    // ---- steady state: t = 1 .. T-PIPE-1 ----
    for (int t = 1; t <= T - PIPE - 1; ++t) {
        wait_async<(PIPE - 1) * 4>();
        consume(t & (PIPE - 1), false);
        wait_ds0();
        issue(t + PIPE, t & (PIPE - 1));
    }

    // ---- epilogue: drain last PIPE slices with constant-immediate waits ----
    {
        const int t0 = T - PIPE;
        wait_async<28>(); consume((t0 + 0) & (PIPE - 1), false);
        wait_async<24>(); consume((t0 + 1) & (PIPE - 1), false);
        wait_async<20>(); consume((t0 + 2) & (PIPE - 1), false);
        wait_async<16>(); consume((t0 + 3) & (PIPE - 1), false);
        wait_async<12>(); consume((t0 + 4) & (PIPE - 1), false);
        wait_async<8>();  consume((t0 + 5) & (PIPE - 1), false);
        wait_async<4>();  consume((t0 + 6) & (PIPE - 1), false);
        wait_async<0>();  consume((t0 + 7) & (PIPE - 1), false);
    }

    // ---- write the 16 outputs per lane (4 x b128) ----
    float* o = out + segBase + (size_t)(lane * 4);
#pragma unroll
    for (int q = 0; q < 4; ++q)
        *reinterpret_cast<float4*>(o + q * 128) = acc[q];
}

extern "C" void kernel_launch(void* const* d_in, const int* in_sizes, int n_in,
                              void* d_out, int out_size, void* d_ws, size_t ws_size,
                              hipStream_t stream)
{
    (void)n_in; (void)d_ws; (void)ws_size;
    const float* adj = (const float*)d_in[0];   // [T, N, N] f32
    const float* lam = (const float*)d_in[1];   // scalar f32
    float*       out = (float*)d_out;           // [N, N] f32

    const int NN = out_size;                    // N*N = 1,048,576
    const int T  = in_sizes[0] / out_size;      // 256

    const int    blocks = NN / BLOCK_FLOATS;    // 256 blocks
    const size_t shmem  = (size_t)WAVES * PIPE * BUF_BYTES;  // 128 KB dynamic LDS

    hipLaunchKernelGGL(ewma_fold_kernel, dim3(blocks), dim3(TBLOCK), shmem, stream,
                       adj, lam, out, T, NN);
}